// VIN_32126355374872
// MI455X (gfx1250) — compile-verified
//
#include <hip/hip_runtime.h>
#include <hip/hip_bf16.h>

// VIN (Value Iteration Network), MI455X / gfx1250.
// Only batch 0 feeds the output; the 150-ch hidden conv folds exactly into an
// effective 2->1 3x3 conv. Everything fits in one WGP's LDS; the 20 serial VI
// steps run inside a single persistent 1024-thread workgroup with the 2->8
// conv expressed as chained V_WMMA_F32_16X16X4_F32 GEMMs (full fp32).

typedef float v2f __attribute__((ext_vector_type(2)));
typedef float v8f __attribute__((ext_vector_type(8)));

#define HID_C   150
#define ACT     8
#define VITERS  20
#define KQ      64
#define HW      64
#define PW      66                 // padded width (zero halo)
#define PAD_AREA (PW * PW)         // 4356

#if defined(__has_builtin)
#if __has_builtin(__builtin_amdgcn_mov_dpp8)
#define VIN_USE_DPP8 1
#endif
#endif

// dpp8 selector immediate for "lane i reads lane (i xor m)" within 8-lane rows
#define DPP8_XOR(m) ( ((0^(m))&7)       | (((1^(m))&7)<<3)  | (((2^(m))&7)<<6)  | \
                      (((3^(m))&7)<<9)  | (((4^(m))&7)<<12) | (((5^(m))&7)<<15) | \
                      (((6^(m))&7)<<18) | (((7^(m))&7)<<21) )

template <int IMM>
static __device__ __forceinline__ float xor_max_step(float x) {
#ifdef VIN_USE_DPP8
    int s = __builtin_amdgcn_mov_dpp8(__float_as_int(x), IMM);
    return fmaxf(x, __int_as_float(s));
#else
    // fallback: lane-xor via shuffle (IMM encodes sel of i^m; recover m from sel of lane 0)
    return fmaxf(x, __shfl_xor(x, IMM & 7, 32));
#endif
}

__global__ __launch_bounds__(1024)
void vin_fused_kernel(const float* __restrict__ X,
                      const int*   __restrict__ S1,
                      const int*   __restrict__ S2,
                      const float* __restrict__ Wh,
                      const float* __restrict__ bh,
                      const float* __restrict__ Wr,
                      const float* __restrict__ Wq,
                      float*       __restrict__ out)
{
    __shared__ float xv[2 * PAD_AREA];   // X staging; reused as v ping-pong
    __shared__ float rpad[PAD_AREA];     // reward map, zero halo
    __shared__ float wf[20];             // folded Weff[18] + beff

    const int tid  = threadIdx.x;
    const int lane = tid & 31;
    const int wave = tid >> 5;           // 0..31
    const int n    = lane & 15;          // WMMA N index (action), also A's M
    const int hi   = lane >> 4;          // lane half selects K pair / M half

    // Warm L2 for X[0] (both channels, 8192 floats) while we zero LDS.
    if (tid < 512) __builtin_prefetch(X + tid * 16, 0, 3);
    if (tid < HID_C) __builtin_prefetch(Wh + tid * 18, 0, 3);

    // ---- zero LDS (halos must be 0 for pad=1 convs) ----
    for (int i = tid; i < 2 * PAD_AREA; i += 1024) xv[i] = 0.f;
    for (int i = tid; i < PAD_AREA;     i += 1024) rpad[i] = 0.f;
    if (tid < 20) wf[tid] = 0.f;
    __syncthreads();

    // ---- stage X[0] (2 channels) into padded LDS ----
    for (int i = tid; i < 2 * HW * HW; i += 1024) {
        int c = i >> 12;            // /4096
        int p = i & 4095;
        int y = p >> 6, x = p & 63;
        xv[c * PAD_AREA + (y + 1) * PW + (x + 1)] = X[i];
    }
    // ---- fold 150-channel hidden layer: Weff = Wr.Wh, beff = Wr.bh ----
    if (tid < 19) {
        float acc = 0.f;
        if (tid < 18) {
            for (int h = 0; h < HID_C; ++h) acc += Wr[h] * Wh[h * 18 + tid];
        } else {
            for (int h = 0; h < HID_C; ++h) acc += Wr[h] * bh[h];
        }
        wf[tid] = acc;
    }
    __syncthreads();

    // ---- r = conv3x3(X0, Weff) + beff ----
    for (int p = tid; p < HW * HW; p += 1024) {
        int y = p >> 6, x = p & 63;
        float acc = wf[18];
#pragma unroll
        for (int c = 0; c < 2; ++c)
#pragma unroll
            for (int dy = 0; dy < 3; ++dy)
#pragma unroll
                for (int dx = 0; dx < 3; ++dx)
                    acc += wf[c * 9 + dy * 3 + dx] *
                           xv[c * PAD_AREA + (y + dy) * PW + (x + dx)];
        rpad[(y + 1) * PW + (x + 1)] = acc;
    }
    __syncthreads();

    // ---- reuse X staging as v ping-pong buffers; zero them ----
    for (int i = tid; i < 2 * PAD_AREA; i += 1024) xv[i] = 0.f;

    // ---- preload B operand (Wq as [K=20 padded][N=16]) : 5 k=4 chunks ----
    // B 4x16 f32 layout: lanes 0-15 hold rows K={0,1} in v2f[0],[1];
    // lanes 16-31 hold rows K={2,3}. N = lane&15.
    v2f Bc[5];
#pragma unroll
    for (int c = 0; c < 5; ++c) {
#pragma unroll
        for (int j = 0; j < 2; ++j) {
            int k = c * 4 + 2 * hi + j;     // rv-channel-major: k = ch*9 + tap
            float w = 0.f;
            if (n < ACT && k < 18) w = Wq[n * 18 + k];
            Bc[c][j] = w;
        }
    }

    // ---- lane-constant A-slot decode, hoisted out of the VI loop ----
    // slot s=(chunk,j): k = 4*(s>>1) + 2*hi + (s&1)
    //   k <  9 : r channel, tap k      (base = rpad)
    //   k < 18 : v channel, tap k-9    (base = vcur)
    //   k >=18 : K padding -> read vcur[0] (halo cell, provably 0.0 forever)
    int doff[10];   // (dy*PW + dx), 0 for padding slots
    int tmsk[10];   // -1 normally, 0 for padding slots (masks tile base to 0)
    int risr[10];   // 1 -> rpad, 0 -> vcur
#pragma unroll
    for (int s = 0; s < 10; ++s) {
        int  k   = (s >> 1) * 4 + 2 * hi + (s & 1);
        bool isr = (k < 9), isz = (k >= 18);
        int  kk  = isr ? k : (k - 9);
        if (isz) kk = 0;
        doff[s] = isz ? 0 : (kk / 3) * PW + (kk % 3);
        tmsk[s] = isz ? 0 : -1;
        risr[s] = isr ? 1 : 0;
    }
    __syncthreads();

    float* vcur = xv;              // v_t   (reads)
    float* vnxt = xv + PAD_AREA;   // v_t+1 (writes)

    // ---- 19 value-iteration updates (20th step's q is only needed at queries)
#pragma unroll 1
    for (int it = 0; it < VITERS - 1; ++it) {
#pragma unroll 2
        for (int t = 0; t < 8; ++t) {
            const int tile = wave + t * 32;          // 256 strips of 1x16 pixels
            const int ty   = tile >> 2;              // row 0..63
            const int tx   = (tile & 3) << 4;        // col base
            const int tb   = ty * PW + tx + n;       // this lane's patch origin

            v8f acc = {};
#pragma unroll
            for (int c = 0; c < 5; ++c) {
                v2f a;
#pragma unroll
                for (int j = 0; j < 2; ++j) {
                    const int s = c * 2 + j;
                    const float* bp  = risr[s] ? rpad : vcur;
                    a[j] = bp[(tb & tmsk[s]) + doff[s]];   // branchless, 1 ds_load
                }
                acc = __builtin_amdgcn_wmma_f32_16x16x4_f32(
                    false, a, false, Bc[c], (short)0, acc, false, false);
            }

            // v[m] = max over 8 actions. Actions live in lanes n=0..7 of each
            // half; DPP8 xor-1/2/4 reduces within contiguous 8-lane rows, so
            // the zero-padded columns (n>=8) never contaminate the max.
            float vred[8];
#pragma unroll
            for (int i = 0; i < 8; ++i) {
                float m = acc[i];
                m = xor_max_step<DPP8_XOR(1)>(m);
                m = xor_max_step<DPP8_XOR(2)>(m);
                m = xor_max_step<DPP8_XOR(4)>(m);
                vred[i] = m;
            }
            if (n == 0) {                            // lanes 0 and 16 store
                int base = (ty + 1) * PW + tx + 1 + 8 * hi;  // M = i + 8*hi
#pragma unroll
                for (int i = 0; i < 8; ++i) vnxt[base + i] = vred[i];
            }
        }
        __syncthreads();
        float* tmp = vcur; vcur = vnxt; vnxt = tmp;
    }
    // vcur now holds v_19; q_20 = conv3x3([r, v_19], Wq), gathered at queries.

    if (tid < KQ) {
        const int y = S1[tid], x = S2[tid];
        float q[ACT];
#pragma unroll
        for (int a = 0; a < ACT; ++a) {
            float acc = 0.f;
#pragma unroll
            for (int dy = 0; dy < 3; ++dy)
#pragma unroll
                for (int dx = 0; dx < 3; ++dx) {
                    int idx = (y + dy) * PW + (x + dx);
                    acc += Wq[a * 18 +     dy * 3 + dx] * rpad[idx];
                    acc += Wq[a * 18 + 9 + dy * 3 + dx] * vcur[idx];
                }
            q[a] = acc;
        }
        float m = q[0];
#pragma unroll
        for (int a = 1; a < ACT; ++a) m = fmaxf(m, q[a]);
        float s = 0.f;
#pragma unroll
        for (int a = 0; a < ACT; ++a) { q[a] = expf(q[a] - m); s += q[a]; }
        float inv = 1.f / s;
#pragma unroll
        for (int a = 0; a < ACT; ++a) out[tid * ACT + a] = q[a] * inv;
    }
}

extern "C" void kernel_launch(void* const* d_in, const int* in_sizes, int n_in,
                              void* d_out, int out_size, void* d_ws, size_t ws_size,
                              hipStream_t stream) {
    const float* X  = (const float*)d_in[0];   // [64,2,64,64]
    const int*   S1 = (const int*)d_in[1];     // [64]
    const int*   S2 = (const int*)d_in[2];     // [64]
    const float* Wh = (const float*)d_in[3];   // [150,2,3,3]
    const float* bh = (const float*)d_in[4];   // [150]
    const float* Wr = (const float*)d_in[5];   // [1,150,1,1]
    const float* Wq = (const float*)d_in[6];   // [8,2,3,3]
    float* out = (float*)d_out;                // [64,8] softmax

    vin_fused_kernel<<<1, 1024, 0, stream>>>(X, S1, S2, Wh, bh, Wr, Wq, out);
}